// CovarianceLayer_65592740544960
// MI455X (gfx1250) — compile-verified
//
#include <hip/hip_runtime.h>

// CDNA5 / gfx1250 covariance kernel: cov[b,c] = Xc^T Xc / (T-1) via
// V_WMMA_F32_16X16X4_F32 Gram accumulation + one-pass mean correction.

typedef __attribute__((ext_vector_type(2))) float v2f;
typedef __attribute__((ext_vector_type(8))) float v8f;

#define B_DIM 64
#define C_DIM 4
#define T_DIM 8192
#define M_DIM 16

__global__ __launch_bounds__(256) void cov_wmma_f32_kernel(const float* __restrict__ x,
                                                           float* __restrict__ out) {
    const int tid  = threadIdx.x;
    const int wave = tid >> 5;      // 0..7
    const int lane = tid & 31;
    const int col  = lane & 15;     // N (and M for A operand) index
    const int half = lane >> 4;     // selects K=0..1 vs K=2..3 slots

    const size_t bc = blockIdx.x;   // 0..255  (b*C + c)
    const float* __restrict__ xbc = x + bc * (size_t)(T_DIM * M_DIM);

    __shared__ float sG[8][256];    // per-wave partial Gram matrices
    __shared__ float sSum[256];     // per-thread partial column sums
    __shared__ float sTot[16];      // reduced column sums

    // Each wave covers 1024 timesteps. One WMMA consumes 4 timesteps.
    // Lane layout for f32 16x16x4 A (16x4) == B (4x16) register image:
    //   lane L, vgpr v  ->  x[t0 + 2*(L/16) + v, L%16]
    const float* p = xbc + (size_t)(wave * 1024 + half * 2) * M_DIM + col;

    v8f c0 = {};
    v8f c1 = {};
    float sum = 0.0f;

    // 128 iterations x 8 timesteps = 1024 timesteps per wave.
#pragma unroll 4
    for (int i = 0; i < 128; ++i) {
        v2f a0, a1;
        a0.x = p[0];                 // row t0 + 2*half
        a0.y = p[M_DIM];             // row t0 + 2*half + 1
        a1.x = p[4 * M_DIM];         // row t0 + 4 + 2*half
        a1.y = p[5 * M_DIM];         // row t0 + 4 + 2*half + 1
        sum += a0.x + a0.y + a1.x + a1.y;
        // D = A*B + C ; A and B use the same register image (Gram matrix).
        c0 = __builtin_amdgcn_wmma_f32_16x16x4_f32(false, a0, false, a0,
                                                   (short)0, c0, false, false);
        c1 = __builtin_amdgcn_wmma_f32_16x16x4_f32(false, a1, false, a1,
                                                   (short)0, c1, false, false);
        p += 8 * M_DIM;
    }
    c0 += c1;

    // C/D layout: vgpr r, lanes<16 -> M=r ; lanes>=16 -> M=r+8 ; N = lane%16.
#pragma unroll
    for (int r = 0; r < 8; ++r) {
        sG[wave][(r + half * 8) * 16 + col] = c0[r];
    }
    sSum[tid] = sum;
    __syncthreads();

    // Reduce column sums: s[m] = sum over all threads whose col == m.
    if (tid < 16) {
        float s = 0.0f;
#pragma unroll
        for (int j = 0; j < 16; ++j) s += sSum[tid + j * 16];
        sTot[tid] = s;
    }
    __syncthreads();

    // Reduce the 8 wave-partial Gram matrices and apply mean correction:
    // cov[m,n] = (G[m,n] - s[m]*s[n]/T) / (T-1)
    float g = 0.0f;
#pragma unroll
    for (int w = 0; w < 8; ++w) g += sG[w][tid];
    const int m = tid >> 4;
    const int n = tid & 15;
    const float invT  = 1.0f / (float)T_DIM;
    const float invT1 = 1.0f / (float)(T_DIM - 1);
    const float cov = (g - sTot[m] * sTot[n] * invT) * invT1;
    out[bc * 256 + tid] = cov;
}

extern "C" void kernel_launch(void* const* d_in, const int* in_sizes, int n_in,
                              void* d_out, int out_size, void* d_ws, size_t ws_size,
                              hipStream_t stream) {
    (void)in_sizes; (void)n_in; (void)out_size; (void)d_ws; (void)ws_size;
    const float* x = (const float*)d_in[0];
    float* out = (float*)d_out;
    dim3 grid(B_DIM * C_DIM);   // one block per (b,c) channel
    dim3 block(256);            // 8 waves of 32
    hipLaunchKernelGGL(cov_wmma_f32_kernel, grid, block, 0, stream, x, out);
}